// DopplerAttention_35450660061767
// MI455X (gfx1250) — compile-verified
//
#include <hip/hip_runtime.h>

// ---------------------------------------------------------------------------
// DopplerAttention on MI455X (gfx1250)
//
// Algebra:  scores = Wq (xr^T xr) Wk^T / 8  =>  per range bin r:
//   G = xr^T xr   (64x64, K=4096)   -- WMMA f32 contraction (only heavy math)
//   attn = softmax(Wq G Wk^T / 8); u = Wv^T colsum(attn)
//   out[s] = xr[s,:] . u            -- coalesced float4 matvec
// Memory bound: 2 passes over 268 MB f32 => ~23 us floor @ 23.3 TB/s.
//
// Double-buffered GLOBAL_LOAD_ASYNC_TO_LDS_B128 staging (ASYNCcnt-tracked)
// so chunk c+1's HBM traffic overlaps chunk c's WMMAs.
// ---------------------------------------------------------------------------

typedef __attribute__((ext_vector_type(2))) float v2f;
typedef __attribute__((ext_vector_type(4))) float v4f;
typedef __attribute__((ext_vector_type(8))) float v8f;
typedef __attribute__((ext_vector_type(4))) int   v4i;

#define A1 __attribute__((address_space(1)))
#define A3 __attribute__((address_space(3)))

#define D_DIM   64
#define R_DIM   256
#define S_DIM   4096
#define CHUNK   128
#define PITCH   132                       // CHUNK + 4 pad floats: conflict-free, 16B rows
#define NCHUNK  (S_DIM / CHUNK)
#define DSTRIDE ((size_t)R_DIM * S_DIM)   // stride between doppler planes of x

#if __has_builtin(__builtin_amdgcn_global_load_async_to_lds_b128)
#define ASYNC_VIA_BUILTIN 1
#else
#define ASYNC_VIA_BUILTIN 0
#endif

template <int N>
__device__ __forceinline__ void wait_asynccnt() {
#if __has_builtin(__builtin_amdgcn_s_wait_asynccnt)
    __builtin_amdgcn_s_wait_asynccnt(N);
#else
    asm volatile("s_wait_asynccnt %0" ::"n"(N) : "memory");
#endif
}

// Issue 8 async b128 copies per thread: A[0:64, s0:s0+128] -> LDS buf.
__device__ __forceinline__ void stage_chunk_async(const float* __restrict__ xr,
                                                  int s0, float* buf, int tid) {
    #pragma unroll
    for (int i = 0; i < 8; ++i) {
        int l  = tid + 256 * i;
        int d  = l >> 5;                  // 32 float4 per 128-float row
        int jv = l & 31;
        const float* g  = xr + (size_t)d * DSTRIDE + (size_t)(s0 + jv * 4);
        float*       l4 = &buf[d * PITCH + jv * 4];
#if ASYNC_VIA_BUILTIN
        __builtin_amdgcn_global_load_async_to_lds_b128((A1 v4i*)g, (A3 v4i*)l4, 0, 0);
#else
        // GV-mode async load: vdst = 32-bit LDS offset (low bits of shared
        // aperture address), vaddr = 64-bit global address, saddr = off.
        asm volatile("global_load_async_to_lds_b128 %0, %1, off"
                     :: "v"((unsigned int)(unsigned long long)(uintptr_t)l4),
                        "v"((unsigned long long)(uintptr_t)g)
                     : "memory");
#endif
    }
}

__global__ __launch_bounds__(256, 1)
void doppler_attn_kernel(const float* __restrict__ x,
                         const float* __restrict__ Wq,
                         const float* __restrict__ Wk,
                         const float* __restrict__ Wv,
                         float* __restrict__ out)
{
    __shared__ float lds_xA[D_DIM * PITCH];  // 33.8 KB staging (ping)
    __shared__ float lds_xB[D_DIM * PITCH];  // 33.8 KB staging (pong)
    __shared__ float lds_G[D_DIM * D_DIM];   // 16 KB Gram matrix
    __shared__ float lds_T[D_DIM * D_DIM];   // 16 KB temp (G @ Wk^T)
    __shared__ float lds_S[D_DIM * D_DIM];   // 16 KB scores / attn
    __shared__ float lds_w[D_DIM];
    __shared__ float lds_u[D_DIM];

    const int r    = blockIdx.x;
    const int tid  = threadIdx.x;
    const int wave = tid >> 5;
    const int lane = tid & 31;
    const int half = lane >> 4;        // K-half per ISA 16x4 f32 A layout
    const int m16  = lane & 15;        // M (and N) index within tile

    const float* xr = x + (size_t)r * S_DIM;   // x[d][r][s] = xr[d*DSTRIDE + s]

    // 16 G-tiles (4x4 of 16x16); 8 waves * 2 tiles; pair shares row-block d1.
    const int ti0 = wave * 2;
    const int d1  = (ti0 >> 2) * 16;
    const int d2a = (ti0 & 3) * 16;
    const int d2b = ((ti0 + 1) & 3) * 16;

    v8f acc0 = {};
    v8f acc1 = {};

    // ---------------- Phase 1: G = xr^T xr via f32 WMMA, async-double-buffered
    stage_chunk_async(xr, 0, lds_xA, tid);            // prologue: chunk 0

    for (int c = 0; c < NCHUNK; ++c) {
        float* cur = (c & 1) ? lds_xB : lds_xA;
        float* nxt = (c & 1) ? lds_xA : lds_xB;

        if (c + 1 < NCHUNK) {
            stage_chunk_async(xr, (c + 1) * CHUNK, nxt, tid);
            wait_asynccnt<8>();    // in-order: 8 newest pending => chunk c done
        } else {
            wait_asynccnt<0>();
        }
        __syncthreads();           // all waves' slices of `cur` visible

        // 32 k-steps of V_WMMA_F32_16X16X4_F32 per tile.
        // A frag (16x4): lane L<16 holds {K=0,1}, L>=16 holds {K=2,3}, row M=L%16.
        // B frag (4x16) mirrors it under Gram symmetry (B[k,n] = A[d2+n, s0+k]).
        #pragma unroll 4
        for (int kk = 0; kk < CHUNK / 4; ++kk) {
            int sb = kk * 4 + 2 * half;
            v2f av = *(const v2f*)&cur[(d1  + m16) * PITCH + sb];
            v2f b0 = *(const v2f*)&cur[(d2a + m16) * PITCH + sb];
            v2f b1 = *(const v2f*)&cur[(d2b + m16) * PITCH + sb];
            acc0 = __builtin_amdgcn_wmma_f32_16x16x4_f32(false, av, false, b0,
                                                         (short)0, acc0, false, false);
            acc1 = __builtin_amdgcn_wmma_f32_16x16x4_f32(false, av, false, b1,
                                                         (short)0, acc1, false, false);
        }
        __syncthreads();           // `cur` free for reuse as the next `nxt`
    }

    // Spill accumulators: C/D layout -> row = v + 8*half, col = lane%16.
    #pragma unroll
    for (int v = 0; v < 8; ++v) {
        lds_G[(d1 + v + 8 * half) * D_DIM + d2a + m16] = acc0[v];
        lds_G[(d1 + v + 8 * half) * D_DIM + d2b + m16] = acc1[v];
    }
    __syncthreads();

    // ---------------- Phase 2: tiny 64x64 math on VALU ----------------
    // 2a: T = G @ Wk^T
    #pragma unroll
    for (int cq = 0; cq < 16; ++cq) {
        int idx = tid + 256 * cq;
        int i = idx >> 6, j = idx & 63;
        float s = 0.f;
        for (int mm = 0; mm < D_DIM; ++mm)
            s = fmaf(lds_G[i * D_DIM + mm], Wk[j * D_DIM + mm], s);
        lds_T[i * D_DIM + j] = s;
    }
    __syncthreads();

    // 2b: scores = (Wq @ T) / 8
    #pragma unroll
    for (int cq = 0; cq < 16; ++cq) {
        int idx = tid + 256 * cq;
        int i = idx >> 6, j = idx & 63;
        float s = 0.f;
        for (int mm = 0; mm < D_DIM; ++mm)
            s = fmaf(Wq[i * D_DIM + mm], lds_T[mm * D_DIM + j], s);
        lds_S[i * D_DIM + j] = 0.125f * s;
    }
    __syncthreads();

    // 2c: row softmax (64 rows, one thread each)
    if (tid < D_DIM) {
        float mx = -1e30f;
        for (int j = 0; j < D_DIM; ++j) mx = fmaxf(mx, lds_S[tid * D_DIM + j]);
        float sum = 0.f;
        for (int j = 0; j < D_DIM; ++j) {
            float e = __expf(lds_S[tid * D_DIM + j] - mx);
            lds_S[tid * D_DIM + j] = e;
            sum += e;
        }
        float inv = 1.f / sum;
        for (int j = 0; j < D_DIM; ++j) lds_S[tid * D_DIM + j] *= inv;
    }
    __syncthreads();

    // 2d: w[e] = sum_d attn[d,e]
    if (tid < D_DIM) {
        float s = 0.f;
        for (int d = 0; d < D_DIM; ++d) s += lds_S[d * D_DIM + tid];
        lds_w[tid] = s;
    }
    __syncthreads();

    // 2e: u[m] = sum_e w[e] * Wv[e,m]
    if (tid < D_DIM) {
        float s = 0.f;
        for (int e = 0; e < D_DIM; ++e) s = fmaf(lds_w[e], Wv[e * D_DIM + tid], s);
        lds_u[tid] = s;
    }
    __syncthreads();

    // ---------------- Phase 3: out[s] = xr[s,:] . u  (float4 coalesced) ------
    v4f oacc[4];
    #pragma unroll
    for (int i = 0; i < 4; ++i) oacc[i] = (v4f){0.f, 0.f, 0.f, 0.f};

    for (int d = 0; d < D_DIM; ++d) {
        float ud = lds_u[d];
        const float* row = xr + (size_t)d * DSTRIDE;
        #pragma unroll
        for (int i = 0; i < 4; ++i) {
            v4f xv = *(const v4f*)(row + tid * 4 + 1024 * i);
            oacc[i] += xv * ud;
        }
    }

    float* orow = out + (size_t)r * S_DIM;
    #pragma unroll
    for (int i = 0; i < 4; ++i)
        *(v4f*)(orow + tid * 4 + 1024 * i) = oacc[i];
}

extern "C" void kernel_launch(void* const* d_in, const int* in_sizes, int n_in,
                              void* d_out, int out_size, void* d_ws, size_t ws_size,
                              hipStream_t stream) {
    (void)in_sizes; (void)n_in; (void)d_ws; (void)ws_size; (void)out_size;
    const float* x  = (const float*)d_in[0];
    const float* Wq = (const float*)d_in[1];
    const float* Wk = (const float*)d_in[2];
    const float* Wv = (const float*)d_in[3];
    float* out = (float*)d_out;
    doppler_attn_kernel<<<R_DIM, 256, 0, stream>>>(x, Wq, Wk, Wv, out);
}